// FCOSPostProcessor_29480655520290
// MI455X (gfx1250) — compile-verified
//
#include <hip/hip_runtime.h>

// ---------------- problem constants ----------------
#define NLVL   5
#define BATCH  8
#define NCLS   16
#define TOPK   1000
#define NCAND  5000      // 5 levels * 1000 per image
#define SORTN  8192      // padded candidate count for bitonic sort
#define FKEEP  112       // 100 padded to 7*16 for WMMA tiling
#define POSTN  100

__constant__ int c_HW[NLVL] = {16000, 4000, 1000, 260, 70};

typedef __attribute__((ext_vector_type(2))) float v2f;
typedef __attribute__((ext_vector_type(8))) float v8f;

// monotonic float <-> uint key (all floats, total order)
__device__ __forceinline__ unsigned f2k(float f) {
  unsigned b = __float_as_uint(f);
  return (b & 0x80000000u) ? ~b : (b | 0x80000000u);
}
__device__ __forceinline__ float k2f(unsigned k) {
  return (k & 0x80000000u) ? __uint_as_float(k & 0x7FFFFFFFu) : __uint_as_float(~k);
}
__device__ __forceinline__ float sigmoidf(float x) { return 1.0f / (1.0f + expf(-x)); }

// ---------------- K1: fused decode of masked scores -> sortable keys ----------------
__global__ void decode_keys_kernel(const float* __restrict__ cls,
                                   const float* __restrict__ ctr,
                                   unsigned* __restrict__ keysOut,
                                   int HW, int HWC, int total) {
  int gid = blockIdx.x * blockDim.x + threadIdx.x;
  if (gid >= total) return;
  int n = gid / HWC;
  int r = gid - n * HWC;      // r = i*16 + c   (matches (N,HW,C) flatten in reference)
  int i = r >> 4;
  int c = r & 15;
  float s1 = sigmoidf(cls[((n * NCLS) + c) * HW + i]);
  float sc = s1 * sigmoidf(ctr[n * HW + i]);
  float masked = (s1 > 0.05f) ? sc : -1.0f;   // PRE_NMS_THRESH
  keysOut[gid] = f2k(masked);
}

// ---------------- K2: exact top-1000 per (image, level) via 4-pass radix select ----------------
__global__ void radix_select_kernel(const unsigned* __restrict__ keysLvl,
                                    unsigned* __restrict__ selIdx,   // pre-offset by l*TOPK
                                    unsigned* __restrict__ selKey,
                                    int HWC) {
  const int n = blockIdx.x;
  const int t = threadIdx.x;
  const unsigned* keys = keysLvl + (size_t)n * (size_t)HWC;
  __shared__ unsigned hist[256];
  __shared__ unsigned sPrefix, sNeed, sCntG, sCntT;
  if (t == 0) { sPrefix = 0u; sNeed = TOPK; sCntG = 0u; sCntT = 0u; }
  __syncthreads();
  for (int pass = 0; pass < 4; ++pass) {
    const int shift = 24 - pass * 8;
    const unsigned prefMask = (pass == 0) ? 0u : (0xFFFFFFFFu << (shift + 8));
    const unsigned pref = sPrefix;
    hist[t] = 0u;
    __syncthreads();
    for (int e = t; e < HWC; e += 256) {
      unsigned key = keys[e];
      if ((key & prefMask) == pref) atomicAdd(&hist[(key >> shift) & 255u], 1u);
    }
    __syncthreads();
    if (t == 0) {
      unsigned running = 0, need = sNeed; int chosen = 0;
      for (int b = 255; b >= 0; --b) {
        unsigned c = hist[b];
        if (running + c >= need) { chosen = b; break; }
        running += c;
      }
      sNeed = need - running;                         // ties still required in chosen bin
      sPrefix = pref | ((unsigned)chosen << shift);
    }
    __syncthreads();
  }
  const unsigned kth  = sPrefix;       // exact value of the 1000th-largest key
  const unsigned need = sNeed;         // #ties of kth to keep
  const unsigned m    = TOPK - need;   // #keys strictly greater
  for (int e = t; e < HWC; e += 256) {
    unsigned key = keys[e];
    if (key > kth) {
      unsigned pos = atomicAdd(&sCntG, 1u);
      if (pos < m) { selIdx[n * NCAND + pos] = (unsigned)e; selKey[n * NCAND + pos] = key; }
    } else if (key == kth) {
      unsigned tp = atomicAdd(&sCntT, 1u);
      if (tp < need) { selIdx[n * NCAND + m + tp] = (unsigned)e; selKey[n * NCAND + m + tp] = key; }
    }
  }
}

// ---------------- K3: decode + clip boxes for selected candidates ----------------
struct DecodeParams {
  const float* loc[NLVL];
  const float* reg[NLVL];
  const unsigned* selIdx;
  const unsigned* selKey;
  const int* imsz;
  float* candBox;
  float* candScore;
  int* candLabel;
  int* candValid;
};

__global__ void decode_boxes_kernel(DecodeParams p) {
  int gid = blockIdx.x * blockDim.x + threadIdx.x;
  if (gid >= BATCH * NCAND) return;
  int n = gid / NCAND;
  int q = gid - n * NCAND;
  int l = q / TOPK;
  unsigned e   = p.selIdx[gid];
  unsigned key = p.selKey[gid];
  int HW = c_HW[l];
  int i = (int)(e >> 4);
  int c = (int)(e & 15u);
  bool valid = (key & 0x80000000u) != 0u;           // candidate passed 0.05 threshold
  float sc = k2f(key);                              // cls*ctr (or -1 for padding)
  float score = valid ? sqrtf(fmaxf(sc, 1e-12f)) : 0.0f;
  const float* loc = p.loc[l];
  const float* reg = p.reg[l];
  float px = loc[2 * i], py = loc[2 * i + 1];
  float r0 = reg[((n * 4) + 0) * HW + i];
  float r1 = reg[((n * 4) + 1) * HW + i];
  float r2 = reg[((n * 4) + 2) * HW + i];
  float r3 = reg[((n * 4) + 3) * HW + i];
  float hmax = (float)(p.imsz[n * 2 + 0] - 1);
  float wmax = (float)(p.imsz[n * 2 + 1] - 1);
  float x1 = fminf(fmaxf(px - r0, 0.0f), wmax);
  float y1 = fminf(fmaxf(py - r1, 0.0f), hmax);
  float x2 = fminf(fmaxf(px + r2, 0.0f), wmax);
  float y2 = fminf(fmaxf(py + r3, 0.0f), hmax);
  float* bp = p.candBox + (size_t)gid * 4;
  bp[0] = x1; bp[1] = y1; bp[2] = x2; bp[3] = y2;
  p.candScore[gid] = score;
  p.candLabel[gid] = c + 1;
  p.candValid[gid] = valid ? 1 : 0;
}

// ---------------- K4: per-image LDS bitonic sort (score desc, index asc) ----------------
__global__ void __launch_bounds__(1024) sort_kernel(const float* __restrict__ candScore,
                                                    const int* __restrict__ candValid,
                                                    int* __restrict__ sortedIdx) {
  const int n = blockIdx.x;
  const int t = threadIdx.x;
  __shared__ unsigned long long sk[SORTN];   // 64 KB of the WGP's 320 KB LDS
  for (int j = t; j < SORTN; j += 1024) {
    unsigned long long key = 0ull;
    if (j < NCAND) {
      float s = candValid[n * NCAND + j] ? candScore[n * NCAND + j] : -1.0f; // masked score
      key = ((unsigned long long)f2k(s) << 13) | (unsigned)(8191 - j);       // idx-asc tiebreak
    }
    sk[j] = key;
  }
  __syncthreads();
  for (int k = 2; k <= SORTN; k <<= 1) {
    for (int jj = k >> 1; jj > 0; jj >>= 1) {
      for (int base = t; base < (SORTN >> 1); base += 1024) {
        int i = ((base & ~(jj - 1)) << 1) | (base & (jj - 1));
        int pgt = i | jj;
        bool desc = (i & k) == 0;
        unsigned long long A = sk[i], B = sk[pgt];
        if (desc ? (A < B) : (A > B)) { sk[i] = B; sk[pgt] = A; }
      }
      __syncthreads();
    }
  }
  for (int j = t; j < NCAND; j += 1024)
    sortedIdx[n * NCAND + j] = 8191 - (int)(sk[j] & 8191u);
}

// ---------------- K5: greedy multi-label NMS (thresh 0.6), boxes held in VGPRs ----------------
__global__ void __launch_bounds__(1024) mlnms_kernel(const float* __restrict__ candBox,
                                                     const int* __restrict__ candLabel,
                                                     const int* __restrict__ candValid,
                                                     const int* __restrict__ sortedIdx,
                                                     int* __restrict__ mlKeep) {
  const int n = blockIdx.x;
  const int t = threadIdx.x;
  __shared__ float piv[2][5];
  __shared__ int pflag[2];
  float x1[5], y1[5], x2[5], y2[5], ar[5];
  bool kp[5];
#pragma unroll
  for (int s = 0; s < 5; ++s) {
    int j = s * 1024 + t;
    if (j < NCAND) {
      int q = sortedIdx[n * NCAND + j];
      const float* bp = candBox + (size_t)(n * NCAND + q) * 4;
      // class-offset boxes: 1280 = clip-max+1; identical keep decisions to max(boxes)+1
      float off = (float)candLabel[n * NCAND + q] * 1280.0f;
      float a = bp[0], b = bp[1], cc = bp[2], d = bp[3];
      x1[s] = a + off; y1[s] = b + off; x2[s] = cc + off; y2[s] = d + off;
      ar[s] = fmaxf(cc - a, 0.0f) * fmaxf(d - b, 0.0f);
      kp[s] = candValid[n * NCAND + q] != 0;
    } else { x1[s] = y1[s] = x2[s] = y2[s] = 0.0f; ar[s] = 0.0f; kp[s] = false; }
  }
  if (t == 0) {
    piv[0][0] = x1[0]; piv[0][1] = y1[0]; piv[0][2] = x2[0]; piv[0][3] = y2[0]; piv[0][4] = ar[0];
    pflag[0] = kp[0] ? 1 : 0;
  }
  __syncthreads();
  for (int i = 0; i < NCAND; ++i) {
    const int sl = i & 1;
    const float px1 = piv[sl][0], py1 = piv[sl][1], px2 = piv[sl][2], py2 = piv[sl][3], pa = piv[sl][4];
    const bool pk = pflag[sl] != 0;
    if (pk) {
#pragma unroll
      for (int s = 0; s < 5; ++s) {
        int j = s * 1024 + t;
        if (j > i && kp[s]) {
          float iw = fminf(px2, x2[s]) - fmaxf(px1, x1[s]);
          float ih = fminf(py2, y2[s]) - fmaxf(py1, y1[s]);
          float inter = fmaxf(iw, 0.0f) * fmaxf(ih, 0.0f);
          float iou = inter / fmaxf(pa + ar[s] - inter, 1e-9f);
          if (iou > 0.6f) kp[s] = false;
        }
      }
    }
    const int ni = i + 1;
    if (ni < NCAND && t == (ni & 1023)) {           // owner publishes next pivot (keep[] final)
      const int s1 = ni >> 10;
      const int ns = ni & 1;
#pragma unroll
      for (int s = 0; s < 5; ++s)
        if (s == s1) {                               // compile-time index: arrays stay in VGPRs
          piv[ns][0] = x1[s]; piv[ns][1] = y1[s]; piv[ns][2] = x2[s]; piv[ns][3] = y2[s];
          piv[ns][4] = ar[s]; pflag[ns] = kp[s] ? 1 : 0;
        }
    }
    __syncthreads();   // single barrier per pivot
  }
#pragma unroll
  for (int s = 0; s < 5; ++s) {
    int j = s * 1024 + t;
    if (j < NCAND) mlKeep[n * NCAND + j] = kp[s] ? 1 : 0;
  }
}

// ---------------- K6: top-100, score>0.5, final NMS 0.9 (WMMA area-sum tiles) ----------------
__global__ void __launch_bounds__(256) final_kernel(const float* __restrict__ candBox,
                                                    const float* __restrict__ candScore,
                                                    const int* __restrict__ candLabel,
                                                    const int* __restrict__ candValid,
                                                    const int* __restrict__ sortedIdx,
                                                    const int* __restrict__ mlKeep,
                                                    float* __restrict__ out) {
  const int n = blockIdx.x;
  const int t = threadIdx.x;
  __shared__ float iouM[FKEEP * FKEEP];
  __shared__ float bx1[FKEEP], by1[FKEEP], bx2[FKEEP], by2[FKEEP], barea[FKEEP], bscore[FKEEP];
  __shared__ int blabel[FKEEP], bkeep[FKEEP];
  if (t == 0) {
    int cnt = 0;
    for (int j = 0; j < NCAND && cnt < POSTN; ++j) {   // sorted survivors == top-100 by score
      if (mlKeep[n * NCAND + j]) {
        int q = sortedIdx[n * NCAND + j];
        const float* bp = candBox + (size_t)(n * NCAND + q) * 4;
        float s = candScore[n * NCAND + q];
        bx1[cnt] = bp[0]; by1[cnt] = bp[1]; bx2[cnt] = bp[2]; by2[cnt] = bp[3];
        barea[cnt] = fmaxf(bp[2] - bp[0], 0.0f) * fmaxf(bp[3] - bp[1], 0.0f);
        bscore[cnt] = s;
        blabel[cnt] = candLabel[n * NCAND + q];
        bkeep[cnt] = (s > 0.5f) ? 1 : 0;               // SCORE_THRESH before final NMS
        ++cnt;
      }
    }
    for (; cnt < FKEEP; ++cnt) {
      bx1[cnt] = by1[cnt] = bx2[cnt] = by2[cnt] = 0.0f;
      barea[cnt] = 0.0f; bscore[cnt] = 0.0f; blabel[cnt] = 0; bkeep[cnt] = 0;
    }
  }
  __syncthreads();
  // 16x16 IoU tiles; pairwise (area_i + area_j) via V_WMMA_F32_16X16X4_F32:
  //   A(16x4) = [area_row, 1, 0, 0], B(4x16) = [1; area_col; 0; 0] -> D = area_i + area_j
  const int wave = t >> 5;
  const int lane = t & 31;
  const bool lo = lane < 16;
  for (int tile = wave; tile < 49; tile += 8) {
    const int tm = (tile / 7) * 16;
    const int tn = (tile % 7) * 16;
    v2f a, b;
    a.x = lo ? barea[tm + lane] : 0.0f;   // A VGPR0: K=0 (rows), lanes>=16 hold K=2 -> 0
    a.y = lo ? 1.0f : 0.0f;               // A VGPR1: K=1 -> 1,   lanes>=16 hold K=3 -> 0
    b.x = lo ? 1.0f : 0.0f;               // B VGPR0: row K=0 -> 1
    b.y = lo ? barea[tn + lane] : 0.0f;   // B VGPR1: row K=1 -> area_col
    v8f c = {};
    v8f d = __builtin_amdgcn_wmma_f32_16x16x4_f32(false, a, false, b, (short)0, c, false, false);
    const int gj = tn + (lo ? lane : lane - 16);
#pragma unroll
    for (int r = 0; r < 8; ++r) {
      const int gi = tm + (lo ? r : r + 8);
      float iw = fminf(bx2[gi], bx2[gj]) - fmaxf(bx1[gi], bx1[gj]);
      float ih = fminf(by2[gi], by2[gj]) - fmaxf(by1[gi], by1[gj]);
      float inter = fmaxf(iw, 0.0f) * fmaxf(ih, 0.0f);
      iouM[gi * FKEEP + gj] = inter / fmaxf(d[r] - inter, 1e-9f);
    }
  }
  __syncthreads();
  if (t == 0) {                                        // tiny greedy pass over 112 entries
    for (int i = 0; i < FKEEP; ++i) {
      if (!bkeep[i]) continue;
      for (int j = i + 1; j < FKEEP; ++j)
        if (bkeep[j] && iouM[i * FKEEP + j] > 0.9f) bkeep[j] = 0;
    }
  }
  __syncthreads();
  for (int r = t; r < POSTN; r += 256) {
    const bool kv = bkeep[r] != 0;
    float* ob = out + ((size_t)n * POSTN + r) * 4;
    ob[0] = kv ? bx1[r] : 0.0f; ob[1] = kv ? by1[r] : 0.0f;
    ob[2] = kv ? bx2[r] : 0.0f; ob[3] = kv ? by2[r] : 0.0f;
    out[BATCH * POSTN * 4 + n * POSTN + r] = kv ? bscore[r] : 0.0f;
    out[BATCH * POSTN * 5 + n * POSTN + r] = kv ? (float)blabel[r] : 0.0f;
    out[BATCH * POSTN * 6 + n * POSTN + r] = kv ? 1.0f : 0.0f;
  }
}

// ---------------- host launcher ----------------
extern "C" void kernel_launch(void* const* d_in, const int* in_sizes, int n_in,
                              void* d_out, int out_size, void* d_ws, size_t ws_size,
                              hipStream_t stream) {
  (void)in_sizes; (void)n_in; (void)out_size; (void)ws_size;
  static const int h_HW[NLVL] = {16000, 4000, 1000, 260, 70};

  const float* loc[NLVL]; const float* cls[NLVL]; const float* reg[NLVL]; const float* ctr[NLVL];
  for (int l = 0; l < NLVL; ++l) {             // setup_inputs() dict order: interleaved per level
    loc[l] = (const float*)d_in[4 * l + 0];
    cls[l] = (const float*)d_in[4 * l + 1];
    reg[l] = (const float*)d_in[4 * l + 2];
    ctr[l] = (const float*)d_in[4 * l + 3];
  }
  const int* imsz = (const int*)d_in[20];
  float* out = (float*)d_out;

  // workspace carve-up (~12.7 MB, all 4B aligned)
  const size_t KEYS_ELEMS = (size_t)NCLS * BATCH * (16000 + 4000 + 1000 + 260 + 70); // 2,730,240
  const size_t SEL_ELEMS  = (size_t)BATCH * NCAND;                                    // 40,000
  unsigned char* w  = (unsigned char*)d_ws;
  unsigned* keys    = (unsigned*)w;
  unsigned* selIdx  = (unsigned*)(w + KEYS_ELEMS * sizeof(unsigned));
  unsigned* selKey  = selIdx + SEL_ELEMS;
  float* candBox    = (float*)(selKey + SEL_ELEMS);
  float* candScore  = candBox + SEL_ELEMS * 4;
  int* candLabel    = (int*)(candScore + SEL_ELEMS);
  int* candValid    = candLabel + SEL_ELEMS;
  int* sortedIdx    = candValid + SEL_ELEMS;
  int* mlKeep       = sortedIdx + SEL_ELEMS;

  // K1: decode masked-score keys (per level)
  size_t kbase = 0;
  for (int l = 0; l < NLVL; ++l) {
    int HWC = h_HW[l] * NCLS;
    int total = BATCH * HWC;
    decode_keys_kernel<<<(total + 255) / 256, 256, 0, stream>>>(cls[l], ctr[l], keys + kbase,
                                                                h_HW[l], HWC, total);
    kbase += (size_t)total;
  }
  // K2: exact top-1000 per (image, level)
  kbase = 0;
  for (int l = 0; l < NLVL; ++l) {
    int HWC = h_HW[l] * NCLS;
    radix_select_kernel<<<BATCH, 256, 0, stream>>>(keys + kbase, selIdx + l * TOPK,
                                                   selKey + l * TOPK, HWC);
    kbase += (size_t)BATCH * HWC;
  }
  // K3: decode + clip boxes
  DecodeParams p;
  for (int l = 0; l < NLVL; ++l) { p.loc[l] = loc[l]; p.reg[l] = reg[l]; }
  p.selIdx = selIdx; p.selKey = selKey; p.imsz = imsz;
  p.candBox = candBox; p.candScore = candScore; p.candLabel = candLabel; p.candValid = candValid;
  decode_boxes_kernel<<<(BATCH * NCAND + 255) / 256, 256, 0, stream>>>(p);
  // K4: per-image sort by (masked score desc, index asc)
  sort_kernel<<<BATCH, 1024, 0, stream>>>(candScore, candValid, sortedIdx);
  // K5: greedy multi-label NMS @0.6
  mlnms_kernel<<<BATCH, 1024, 0, stream>>>(candBox, candLabel, candValid, sortedIdx, mlKeep);
  // K6: top-100 + score>0.5 + final NMS @0.9 (WMMA area-sum tiles) + output write
  final_kernel<<<BATCH, 256, 0, stream>>>(candBox, candScore, candLabel, candValid,
                                          sortedIdx, mlKeep, out);
}